// GAT_Modeli_86655260164153
// MI455X (gfx1250) — compile-verified
//
#include <hip/hip_runtime.h>
#include <hip/hip_bf16.h>
#include <math.h>

typedef float v2f __attribute__((ext_vector_type(2)));
typedef float v8f __attribute__((ext_vector_type(8)));

#define NEG_SLOPE 0.2f

__device__ __forceinline__ float leaky(float v) { return v > 0.f ? v : NEG_SLOPE * v; }

// Exact float atomic max via signed/unsigned integer ordering trick.
__device__ __forceinline__ void atomicMaxF(float* addr, float val) {
  if (val >= 0.f) atomicMax((int*)addr, __float_as_int(val));
  else            atomicMin((unsigned int*)addr, (unsigned int)__float_as_int(val));
}

// ---------------- init: zero/fill all accumulators (runs every call) --------
__global__ void init_kern(float* emax1, float* denom1, float* agg1,
                          float* emax2, float* denom2, float* agg2, int N) {
  int i = blockIdx.x * blockDim.x + threadIdx.x;
  if (i < N * 64) agg1[i] = 0.f;
  if (i < N * 6)  agg2[i] = 0.f;
  if (i < N * 4) { emax1[i] = -3.0e38f; denom1[i] = 0.f; }
  if (i < N)     { emax2[i] = -3.0e38f; denom2[i] = 0.f; }
}

// ---------------- layer 1 projection: h1 = x @ W1  (WMMA f32 16x16x4) ------
// One wave -> 16-node tile x 64 cols. K = 7 padded to 8 (two K=4 steps).
__global__ void proj1_wmma(const float* __restrict__ x, const float* __restrict__ W1,
                           float* __restrict__ h1, int N) {
  int gtid = blockIdx.x * blockDim.x + threadIdx.x;
  int wave = gtid >> 5;
  int lane = threadIdx.x & 31;
  int tiles = (N + 15) >> 4;
  if (wave >= tiles) return;           // wave-uniform: EXEC stays all-ones
  int half = lane >> 4;                // K-pair selector for A/B fragments
  int l16  = lane & 15;
  int row  = wave * 16 + l16;          // A: M = lane%16
  int rowc = row < N ? row : N - 1;
  bool rok = row < N;

  v2f a[2];
#pragma unroll
  for (int s = 0; s < 2; ++s) {        // K blocks 0..3 and 4..7
    int k0 = 4 * s + 2 * half;
    a[s][0] = (rok && k0     < 7) ? x[rowc * 7 + k0]     : 0.f;
    a[s][1] = (rok && k0 + 1 < 7) ? x[rowc * 7 + k0 + 1] : 0.f;
  }

  v8f acc[4];
#pragma unroll
  for (int t = 0; t < 4; ++t)
#pragma unroll
    for (int q = 0; q < 8; ++q) acc[t][q] = 0.f;

#pragma unroll
  for (int s = 0; s < 2; ++s) {
    int k0 = 4 * s + 2 * half;
#pragma unroll
    for (int t = 0; t < 4; ++t) {      // 4 column tiles of 16 (64 cols total)
      int col = t * 16 + l16;          // B: N = lane%16
      v2f b;
      b[0] = (k0     < 7) ? W1[k0 * 64 + col]       : 0.f;
      b[1] = (k0 + 1 < 7) ? W1[(k0 + 1) * 64 + col] : 0.f;
      acc[t] = __builtin_amdgcn_wmma_f32_16x16x4_f32(
          false, a[s], false, b, (short)0, acc[t], false, false);
    }
  }

  // D layout: VGPR r -> row r + 8*half, col = t*16 + lane%16
#pragma unroll
  for (int t = 0; t < 4; ++t)
#pragma unroll
    for (int r = 0; r < 8; ++r) {
      int m = wave * 16 + r + 8 * half;
      if (m < N) h1[m * 64 + t * 16 + l16] = acc[t][r];
    }
}

// ---------------- per-(node,head) attention coefficients -------------------
__global__ void alpha1_kern(const float* __restrict__ h1, const float* __restrict__ aS,
                            const float* __restrict__ aD, float* __restrict__ as1,
                            float* __restrict__ ad1, int N) {
  int i = blockIdx.x * blockDim.x + threadIdx.x;
  if (i >= N * 4) return;
  int hd = i & 3, n = i >> 2;
  const float* hp = h1 + n * 64 + hd * 16;
  float s = 0.f, d = 0.f;
#pragma unroll
  for (int c = 0; c < 16; ++c) {
    float v = hp[c];
    s += v * aS[hd * 16 + c];
    d += v * aD[hd * 16 + c];
  }
  as1[i] = s; ad1[i] = d;
}

// ---------------- layer 1 edge passes (incl. implicit self-loops) ----------
__global__ void e1_max(const int* __restrict__ ei, const float* __restrict__ as1,
                       const float* __restrict__ ad1, float* __restrict__ emax1,
                       int E, int N) {
  int e = blockIdx.x * blockDim.x + threadIdx.x;
  if (e >= E + N) return;
  int src = e < E ? ei[e]     : e - E;
  int dst = e < E ? ei[E + e] : e - E;
#pragma unroll
  for (int h = 0; h < 4; ++h) {
    float ev = leaky(as1[src * 4 + h] + ad1[dst * 4 + h]);
    atomicMaxF(&emax1[dst * 4 + h], ev);
  }
}

__global__ void e1_den(const int* __restrict__ ei, const float* __restrict__ as1,
                       const float* __restrict__ ad1, const float* __restrict__ emax1,
                       float* __restrict__ denom1, int E, int N) {
  int e = blockIdx.x * blockDim.x + threadIdx.x;
  if (e >= E + N) return;
  int src = e < E ? ei[e]     : e - E;
  int dst = e < E ? ei[E + e] : e - E;
#pragma unroll
  for (int h = 0; h < 4; ++h) {
    float ev = leaky(as1[src * 4 + h] + ad1[dst * 4 + h]);
    atomicAdd(&denom1[dst * 4 + h], __expf(ev - emax1[dst * 4 + h]));
  }
}

__global__ void e1_agg(const int* __restrict__ ei, const float* __restrict__ as1,
                       const float* __restrict__ ad1, const float* __restrict__ emax1,
                       const float* __restrict__ denom1, const float* __restrict__ h1,
                       float* __restrict__ agg1, int E, int N) {
  int t = blockIdx.x * blockDim.x + threadIdx.x;
  if (t >= (E + N) * 4) return;
  int e = t >> 2, h = t & 3;
  int src = e < E ? ei[e]     : e - E;
  int dst = e < E ? ei[E + e] : e - E;
  float ev = leaky(as1[src * 4 + h] + ad1[dst * 4 + h]);
  float w  = __expf(ev - emax1[dst * 4 + h]) / denom1[dst * 4 + h];
  const float4* hp = (const float4*)(h1 + src * 64 + h * 16);
  float* op = agg1 + dst * 64 + h * 16;
#pragma unroll
  for (int q = 0; q < 4; ++q) {
    float4 v = hp[q];
    atomicAdd(op + q * 4 + 0, v.x * w);
    atomicAdd(op + q * 4 + 1, v.y * w);
    atomicAdd(op + q * 4 + 2, v.z * w);
    atomicAdd(op + q * 4 + 3, v.w * w);
  }
}

// ---------------- bias + ELU (in place: agg1 -> x2) ------------------------
__global__ void bias_elu(float* __restrict__ x2, const float* __restrict__ b1, int N) {
  int i = blockIdx.x * blockDim.x + threadIdx.x;
  if (i >= N * 64) return;
  float v = x2[i] + b1[i & 63];
  x2[i] = v > 0.f ? v : (__expf(v) - 1.f);
}

// ---------------- layer 2 projection: h2 = x2 @ W2 (WMMA, K=64, 6 cols) ----
__global__ void proj2_wmma(const float* __restrict__ x2, const float* __restrict__ W2,
                           float* __restrict__ h2, int N) {
  int gtid = blockIdx.x * blockDim.x + threadIdx.x;
  int wave = gtid >> 5;
  int lane = threadIdx.x & 31;
  int tiles = (N + 15) >> 4;
  if (wave >= tiles) return;
  int half = lane >> 4, l16 = lane & 15;
  int row  = wave * 16 + l16;
  int rowc = row < N ? row : N - 1;
  bool rok = row < N;
  bool cok = l16 < 6;

  v8f acc;
#pragma unroll
  for (int q = 0; q < 8; ++q) acc[q] = 0.f;

#pragma unroll
  for (int s = 0; s < 16; ++s) {       // K = 64 in steps of 4
    int k0 = 4 * s + 2 * half;
    v2f a, b;
    a[0] = rok ? x2[rowc * 64 + k0]     : 0.f;
    a[1] = rok ? x2[rowc * 64 + k0 + 1] : 0.f;
    b[0] = cok ? W2[k0 * 6 + l16]       : 0.f;   // cols 6..15 zero-padded
    b[1] = cok ? W2[(k0 + 1) * 6 + l16] : 0.f;
    acc = __builtin_amdgcn_wmma_f32_16x16x4_f32(
        false, a, false, b, (short)0, acc, false, false);
  }

  if (cok) {
#pragma unroll
    for (int r = 0; r < 8; ++r) {
      int m = wave * 16 + r + 8 * half;
      if (m < N) h2[m * 6 + l16] = acc[r];
    }
  }
}

__global__ void alpha2_kern(const float* __restrict__ h2, const float* __restrict__ aS,
                            const float* __restrict__ aD, float* __restrict__ as2,
                            float* __restrict__ ad2, int N) {
  int n = blockIdx.x * blockDim.x + threadIdx.x;
  if (n >= N) return;
  float s = 0.f, d = 0.f;
#pragma unroll
  for (int c = 0; c < 6; ++c) {
    float v = h2[n * 6 + c];
    s += v * aS[c];
    d += v * aD[c];
  }
  as2[n] = s; ad2[n] = d;
}

// ---------------- layer 2 edge passes (1 head, 6 channels) -----------------
__global__ void e2_max(const int* __restrict__ ei, const float* __restrict__ as2,
                       const float* __restrict__ ad2, float* __restrict__ emax2,
                       int E, int N) {
  int e = blockIdx.x * blockDim.x + threadIdx.x;
  if (e >= E + N) return;
  int src = e < E ? ei[e]     : e - E;
  int dst = e < E ? ei[E + e] : e - E;
  atomicMaxF(&emax2[dst], leaky(as2[src] + ad2[dst]));
}

__global__ void e2_den(const int* __restrict__ ei, const float* __restrict__ as2,
                       const float* __restrict__ ad2, const float* __restrict__ emax2,
                       float* __restrict__ denom2, int E, int N) {
  int e = blockIdx.x * blockDim.x + threadIdx.x;
  if (e >= E + N) return;
  int src = e < E ? ei[e]     : e - E;
  int dst = e < E ? ei[E + e] : e - E;
  float ev = leaky(as2[src] + ad2[dst]);
  atomicAdd(&denom2[dst], __expf(ev - emax2[dst]));
}

__global__ void e2_agg(const int* __restrict__ ei, const float* __restrict__ as2,
                       const float* __restrict__ ad2, const float* __restrict__ emax2,
                       const float* __restrict__ denom2, const float* __restrict__ h2,
                       float* __restrict__ agg2, int E, int N) {
  int e = blockIdx.x * blockDim.x + threadIdx.x;
  if (e >= E + N) return;
  int src = e < E ? ei[e]     : e - E;
  int dst = e < E ? ei[E + e] : e - E;
  float ev = leaky(as2[src] + ad2[dst]);
  float w  = __expf(ev - emax2[dst]) / denom2[dst];
#pragma unroll
  for (int c = 0; c < 6; ++c)
    atomicAdd(&agg2[dst * 6 + c], h2[src * 6 + c] * w);
}

// ---------------- bias + log_softmax ---------------------------------------
__global__ void final_kern(const float* __restrict__ agg2, const float* __restrict__ b2,
                           float* __restrict__ out, int N) {
  int n = blockIdx.x * blockDim.x + threadIdx.x;
  if (n >= N) return;
  float o[6];
  float m = -3.0e38f;
#pragma unroll
  for (int c = 0; c < 6; ++c) { o[c] = agg2[n * 6 + c] + b2[c]; m = fmaxf(m, o[c]); }
  float sum = 0.f;
#pragma unroll
  for (int c = 0; c < 6; ++c) sum += __expf(o[c] - m);
  float l = __logf(sum);
#pragma unroll
  for (int c = 0; c < 6; ++c) out[n * 6 + c] = o[c] - m - l;
}

extern "C" void kernel_launch(void* const* d_in, const int* in_sizes, int n_in,
                              void* d_out, int out_size, void* d_ws, size_t ws_size,
                              hipStream_t stream) {
  const float* x      = (const float*)d_in[0];
  const int*   ei     = (const int*)d_in[1];
  const float* W1     = (const float*)d_in[2];
  const float* a_src1 = (const float*)d_in[3];
  const float* a_dst1 = (const float*)d_in[4];
  const float* b1     = (const float*)d_in[5];
  const float* W2     = (const float*)d_in[6];
  const float* a_src2 = (const float*)d_in[7];
  const float* a_dst2 = (const float*)d_in[8];
  const float* b2     = (const float*)d_in[9];
  float* out = (float*)d_out;

  int N = in_sizes[0] / 7;
  int E = in_sizes[1] / 2;
  int ET = E + N;

  // workspace layout (floats): total N*160 = 64 MB for N=100k
  float* ws     = (float*)d_ws;
  float* h1     = ws;                        // N*64
  float* x2     = h1 + (size_t)N * 64;       // N*64: agg1 then ELU in place
  float* as1    = x2 + (size_t)N * 64;       // N*4
  float* ad1    = as1 + (size_t)N * 4;       // N*4
  float* emax1  = ad1 + (size_t)N * 4;       // N*4
  float* denom1 = emax1 + (size_t)N * 4;     // N*4
  float* h2     = denom1 + (size_t)N * 4;    // N*6
  float* agg2   = h2 + (size_t)N * 6;        // N*6
  float* as2    = agg2 + (size_t)N * 6;      // N
  float* ad2    = as2 + N;                   // N
  float* emax2  = ad2 + N;                   // N
  float* denom2 = emax2 + N;                 // N

  const int B = 256;
  auto G = [](long n, int b) { return (int)((n + b - 1) / b); };
  int tiles = (N + 15) / 16;
  int projBlocks = (tiles + (B / 32) - 1) / (B / 32);

  init_kern<<<G((long)N * 64, B), B, 0, stream>>>(emax1, denom1, x2, emax2, denom2, agg2, N);

  // layer 1
  proj1_wmma<<<projBlocks, B, 0, stream>>>(x, W1, h1, N);
  alpha1_kern<<<G((long)N * 4, B), B, 0, stream>>>(h1, a_src1, a_dst1, as1, ad1, N);
  e1_max<<<G(ET, B), B, 0, stream>>>(ei, as1, ad1, emax1, E, N);
  e1_den<<<G(ET, B), B, 0, stream>>>(ei, as1, ad1, emax1, denom1, E, N);
  e1_agg<<<G((long)ET * 4, B), B, 0, stream>>>(ei, as1, ad1, emax1, denom1, h1, x2, E, N);
  bias_elu<<<G((long)N * 64, B), B, 0, stream>>>(x2, b1, N);

  // layer 2
  proj2_wmma<<<projBlocks, B, 0, stream>>>(x2, W2, h2, N);
  alpha2_kern<<<G(N, B), B, 0, stream>>>(h2, a_src2, a_dst2, as2, ad2, N);
  e2_max<<<G(ET, B), B, 0, stream>>>(ei, as2, ad2, emax2, E, N);
  e2_den<<<G(ET, B), B, 0, stream>>>(ei, as2, ad2, emax2, denom2, E, N);
  e2_agg<<<G(ET, B), B, 0, stream>>>(ei, as2, ad2, emax2, denom2, h2, agg2, E, N);
  final_kern<<<G(N, B), B, 0, stream>>>(agg2, b2, out, N);
}